// DecoderNSE_9053791060380
// MI455X (gfx1250) — compile-verified
//
#include <hip/hip_runtime.h>
#include <math.h>

// Problem dims (fixed by reference)
#define TSTEPS 16
#define B      64
#define H      256
#define N      2048

typedef float v2f  __attribute__((ext_vector_type(2)));
typedef float v8f  __attribute__((ext_vector_type(8)));
typedef unsigned int u32x4 __attribute__((ext_vector_type(4)));
typedef int   i32x4 __attribute__((ext_vector_type(4)));
typedef int   i32x8 __attribute__((ext_vector_type(8)));

__device__ __forceinline__ v8f wmma_f32_16x16x4(v2f a, v2f b, v8f c) {
  return __builtin_amdgcn_wmma_f32_16x16x4_f32(false, a, false, b, (short)0, c,
                                               false, false);
}

__device__ __forceinline__ float sigmoidf_(float x) {
  return 1.0f / (1.0f + __expf(-x));
}

// ---------------------------------------------------------------------------
// TDM staged copy: contiguous run of f32 from global -> LDS, described as a
// 1-row 2D tile (tile_dim0 = nelem, tile_dim1 = 1).  See CDNA5 ISA.
// ---------------------------------------------------------------------------
#if __has_builtin(__builtin_amdgcn_tensor_load_to_lds)
#define HAVE_TDM 1
#else
#define HAVE_TDM 0
#endif

#if __has_include(<hip/amd_detail/amd_gfx1250_TDM.h>)
#define TDM_SIX_ARGS 1
#else
#define TDM_SIX_ARGS 0
#endif

#if HAVE_TDM
__device__ __forceinline__ void tdm_copy_f32_to_lds(const float* gsrc,
                                                    unsigned lds_byte_addr,
                                                    unsigned nelem)
{
  const unsigned long long ga = (unsigned long long)(uintptr_t)gsrc;
  u32x4 g0;
  g0[0] = 1u;                                         // count=1 (valid user D#)
  g0[1] = lds_byte_addr;                              // LDS byte address
  g0[2] = (unsigned)(ga & 0xffffffffu);               // global_addr[31:0]
  g0[3] = (unsigned)((ga >> 32) & 0x01ffffffu)        // global_addr[56:32]
          | (2u << 30);                               // type = 2 ("image")
  i32x8 g1;
  g1[0] = (int)(2u << 16);                            // data_size=2 -> 4 bytes
  g1[1] = (int)((nelem & 0xffffu) << 16);             // tensor_dim0[15:0]
  g1[2] = (int)(((nelem >> 16) & 0xffffu))            // tensor_dim0[31:16]
          | (1 << 16);                                // tensor_dim1 = 1
  g1[3] = (int)((nelem & 0xffffu) << 16);             // tile_dim0
  g1[4] = 1;                                          // tile_dim1=1, tile_dim2=0
  g1[5] = (int)nelem;                                 // tensor_dim0_stride[31:0]
  g1[6] = 0;
  g1[7] = 0;
  const i32x4 zero4 = (i32x4)(0);
#if TDM_SIX_ARGS
  const i32x8 zero8 = (i32x8)(0);
  __builtin_amdgcn_tensor_load_to_lds(g0, g1, zero4, zero4, zero8, 0);
#else
  __builtin_amdgcn_tensor_load_to_lds(g0, g1, zero4, zero4, 0);
#endif
#if __has_builtin(__builtin_amdgcn_s_wait_tensorcnt)
  __builtin_amdgcn_s_wait_tensorcnt(0);
#else
  asm volatile("s_wait_tensorcnt 0x0" ::: "memory");
#endif
}
#endif // HAVE_TDM

// ---------------------------------------------------------------------------
// LSTM cell: gates = X@Wi^T + bi + Hin@Wh^T + bh ; fused activations + state.
// grid = (B/16, H/16), block = 32 (one wave); v_wmma_f32_16x16x4_f32 tiles.
// ---------------------------------------------------------------------------
__global__ __launch_bounds__(32)
void lstm_cell_kernel(const float* __restrict__ X,    // [B,H]
                      const float* __restrict__ Wi,   // [4H,H]
                      const float* __restrict__ Wh,   // [4H,H]
                      const float* __restrict__ Bi,   // [4H]
                      const float* __restrict__ Bh,   // [4H]
                      const float* __restrict__ Hin,  // [B,H]
                      float* __restrict__ Cs,         // [B,H] in/out
                      float* __restrict__ Hout,       // [B,H]
                      float* __restrict__ OutOpt)     // [B,H] or null
{
  const int lane  = threadIdx.x;
  const int mb    = blockIdx.x * 16;
  const int nb    = blockIdx.y * 16;
  const int row16 = lane & 15;
  const int khalf = (lane >> 4) * 2;

  v8f acc[4];
  #pragma unroll
  for (int g = 0; g < 4; ++g) acc[g] = (v8f)(0.0f);

  const float* xrow = X   + (size_t)(mb + row16) * H;
  const float* hrow = Hin + (size_t)(mb + row16) * H;

  for (int k = 0; k < H; k += 4) {
    const int kk = k + khalf;               // even -> 8B aligned
    v2f aX = *(const v2f*)(xrow + kk);
    v2f aH = *(const v2f*)(hrow + kk);
    #pragma unroll
    for (int g = 0; g < 4; ++g) {
      const int wr = (g * H + nb + row16);  // weight row = output column
      v2f bI = *(const v2f*)(Wi + (size_t)wr * H + kk);
      v2f bW = *(const v2f*)(Wh + (size_t)wr * H + kk);
      acc[g] = wmma_f32_16x16x4(aX, bI, acc[g]);
      acc[g] = wmma_f32_16x16x4(aH, bW, acc[g]);
    }
  }

  const int n   = nb + (lane & 15);
  const int mhi = (lane >> 4) * 8;
  float bsum[4];
  #pragma unroll
  for (int g = 0; g < 4; ++g) bsum[g] = Bi[g * H + n] + Bh[g * H + n];

  #pragma unroll
  for (int r = 0; r < 8; ++r) {
    const size_t idx = (size_t)(mb + mhi + r) * H + n;
    const float gi = acc[0][r] + bsum[0];
    const float gf = acc[1][r] + bsum[1];
    const float gg = acc[2][r] + bsum[2];
    const float go = acc[3][r] + bsum[3];
    const float c_new = sigmoidf_(gf) * Cs[idx] + sigmoidf_(gi) * tanhf(gg);
    const float h_new = sigmoidf_(go) * tanhf(c_new);
    Cs[idx]   = c_new;
    Hout[idx] = h_new;
    if (OutOpt) OutOpt[idx] = h_new;
  }
}

// ---------------------------------------------------------------------------
// Attention partial scores: Zp[b,c,n] = sum_{d in chunk c} h[b,d]*mem[b,d,n]
// grid = (B, nchunk), block = 256.  b128 streaming of the mem slab.
// Thread t owns n in {4t..4t+3} and {1024+4t..1024+4t+3}.
// ---------------------------------------------------------------------------
__global__ __launch_bounds__(256)
void attn_partial_kernel(const float* __restrict__ Hr,   // [B,H]
                         const float* __restrict__ Mem,  // [B,H,N]
                         float* __restrict__ Zp,         // [B,nchunk,N]
                         int nchunk)
{
  __shared__ float sh[H];
  const int b = blockIdx.x, c = blockIdx.y, t = threadIdx.x;
  const int rows = H / nchunk;
  const int d0 = c * rows;
  for (int i = t; i < rows; i += 256) sh[i] = Hr[(size_t)b * H + d0 + i];
  __syncthreads();

  const float* memb = Mem + (size_t)b * H * N + (size_t)d0 * N;
  const int n0 = t * 4, n1 = 1024 + t * 4;
  float4 a0 = make_float4(0.f, 0.f, 0.f, 0.f);
  float4 a1 = a0;

  for (int d = 0; d < rows; ++d) {
    const float hv = sh[d];
    const float* row = memb + (size_t)d * N;
    __builtin_prefetch(row + N, 0, 1);
    const float4 r0 = *(const float4*)(row + n0);
    const float4 r1 = *(const float4*)(row + n1);
    a0.x = fmaf(hv, r0.x, a0.x); a0.y = fmaf(hv, r0.y, a0.y);
    a0.z = fmaf(hv, r0.z, a0.z); a0.w = fmaf(hv, r0.w, a0.w);
    a1.x = fmaf(hv, r1.x, a1.x); a1.y = fmaf(hv, r1.y, a1.y);
    a1.z = fmaf(hv, r1.z, a1.z); a1.w = fmaf(hv, r1.w, a1.w);
  }
  float* zp = Zp + ((size_t)b * nchunk + c) * N;
  *(float4*)(zp + n0) = a0;
  *(float4*)(zp + n1) = a1;
}

// ---------------------------------------------------------------------------
// Attention finalize: sum partials, softmax over N, write Z.
// grid = B, block = 256.
// ---------------------------------------------------------------------------
__global__ __launch_bounds__(256)
void attn_finalize_kernel(const float* __restrict__ Zp,  // [B,nchunk,N]
                          float* __restrict__ Z,         // [B,N]
                          int nchunk)
{
  __shared__ float red[16];
  const int b = blockIdx.x, t = threadIdx.x;
  const int n0 = t * 4, n1 = 1024 + t * 4;

  float4 a0 = make_float4(0.f, 0.f, 0.f, 0.f);
  float4 a1 = a0;
  for (int c = 0; c < nchunk; ++c) {
    const float* zp = Zp + ((size_t)b * nchunk + c) * N;
    const float4 p0 = *(const float4*)(zp + n0);
    const float4 p1 = *(const float4*)(zp + n1);
    a0.x += p0.x; a0.y += p0.y; a0.z += p0.z; a0.w += p0.w;
    a1.x += p1.x; a1.y += p1.y; a1.z += p1.z; a1.w += p1.w;
  }

  float v[8] = {a0.x, a0.y, a0.z, a0.w, a1.x, a1.y, a1.z, a1.w};

  // block max
  float mx = v[0];
  #pragma unroll
  for (int j = 1; j < 8; ++j) mx = fmaxf(mx, v[j]);
  for (int off = 16; off > 0; off >>= 1) mx = fmaxf(mx, __shfl_down(mx, off));
  if ((t & 31) == 0) red[t >> 5] = mx;
  __syncthreads();
  if (t == 0) {
    float m2 = red[0];
    for (int i = 1; i < 8; ++i) m2 = fmaxf(m2, red[i]);
    red[0] = m2;
  }
  __syncthreads();
  mx = red[0];

  // exp + block sum
  float s = 0.0f;
  #pragma unroll
  for (int j = 0; j < 8; ++j) { v[j] = __expf(v[j] - mx); s += v[j]; }
  for (int off = 16; off > 0; off >>= 1) s += __shfl_down(s, off);
  if ((t & 31) == 0) red[8 + (t >> 5)] = s;
  __syncthreads();
  if (t == 0) {
    float ss = 0.0f;
    for (int i = 0; i < 8; ++i) ss += red[8 + i];
    red[8] = ss;
  }
  __syncthreads();
  const float inv = 1.0f / red[8];

  float* zb = Z + (size_t)b * N;
  *(float4*)(zb + n0) = make_float4(v[0] * inv, v[1] * inv, v[2] * inv, v[3] * inv);
  *(float4*)(zb + n1) = make_float4(v[4] * inv, v[5] * inv, v[6] * inv, v[7] * inv);
}

// ---------------------------------------------------------------------------
// Retrieve: m[b,d] = sum_n mem[b,d,n] * z[b,n].
// grid = (B, 4), block = 256 (8 waves); wave handles 8 rows of its 64-row
// chunk.  z staged to LDS by the Tensor Data Mover when available.
// ---------------------------------------------------------------------------
__global__ __launch_bounds__(256)
void retrieve_kernel(const float* __restrict__ Mem,  // [B,H,N]
                     const float* __restrict__ Z,    // [B,N]
                     float* __restrict__ Mret)       // [B,H]
{
  __shared__ float zsh[N];   // 8 KB
  const int b = blockIdx.x, c = blockIdx.y, t = threadIdx.x;
  const int wave = t >> 5, lane = t & 31;

#if HAVE_TDM
  if (t < 32) {
    // low 32 bits of a generic LDS pointer == LDS byte address (ISA §10.2)
    tdm_copy_f32_to_lds(Z + (size_t)b * N,
                        (unsigned)(uintptr_t)&zsh[0], N);
  }
#else
  for (int i = t; i < N; i += 256) zsh[i] = Z[(size_t)b * N + i];
#endif
  __syncthreads();

  const int d0 = c * 64;
  const float* memb = Mem + (size_t)b * H * N;
  for (int dr = wave; dr < 64; dr += 8) {
    const float* row = memb + (size_t)(d0 + dr) * N;
    float s = 0.0f;
    #pragma unroll 4
    for (int i = 0; i < N / 128; ++i) {        // 16 iters of b128
      const int n = i * 128 + lane * 4;
      const float4 r  = *(const float4*)(row + n);
      const float4 zz = *(const float4*)(zsh + n);
      s = fmaf(r.x, zz.x, s); s = fmaf(r.y, zz.y, s);
      s = fmaf(r.z, zz.z, s); s = fmaf(r.w, zz.w, s);
    }
    for (int off = 16; off > 0; off >>= 1) s += __shfl_down(s, off);
    if (lane == 0) Mret[(size_t)b * H + d0 + dr] = s;
  }
}

// ---------------------------------------------------------------------------
// Compose: c_t = interleave(h_r, m) @ Wc^T + bc.  kk always even:
// cmp_in[m, kk] = h_r[m, kk/2], cmp_in[m, kk+1] = m[m, kk/2].
// ---------------------------------------------------------------------------
__global__ __launch_bounds__(32)
void compose_kernel(const float* __restrict__ Hr,    // [B,H]
                    const float* __restrict__ Mret,  // [B,H]
                    const float* __restrict__ Wc,    // [H,2H]
                    const float* __restrict__ Bc,    // [H]
                    float* __restrict__ Ct)          // [B,H]
{
  const int lane  = threadIdx.x;
  const int mb    = blockIdx.x * 16;
  const int nb    = blockIdx.y * 16;
  const int row16 = lane & 15;
  const int khalf = (lane >> 4) * 2;

  v8f acc = (v8f)(0.0f);
  const float* hrow = Hr   + (size_t)(mb + row16) * H;
  const float* mrow = Mret + (size_t)(mb + row16) * H;
  const float* wrow = Wc   + (size_t)(nb + row16) * (2 * H);

  for (int k = 0; k < 2 * H; k += 4) {
    const int kk = k + khalf;
    v2f a;
    a.x = hrow[kk >> 1];
    a.y = mrow[kk >> 1];
    v2f bf = *(const v2f*)(wrow + kk);
    acc = wmma_f32_16x16x4(a, bf, acc);
  }

  const int n   = nb + (lane & 15);
  const int mhi = (lane >> 4) * 8;
  const float bias = Bc[n];
  #pragma unroll
  for (int r = 0; r < 8; ++r)
    Ct[(size_t)(mb + mhi + r) * H + n] = acc[r] + bias;
}

// ---------------------------------------------------------------------------
// Memory erase+write: mem = mem*(1-z) + h_w (outer) z.  b128 streaming.
// ---------------------------------------------------------------------------
__global__ __launch_bounds__(256)
void mem_update_kernel(float* __restrict__ Mem,      // [B,H,N]
                       const float* __restrict__ Z,  // [B,N]
                       const float* __restrict__ Hw) // [B,H]
{
  const size_t idx = ((size_t)blockIdx.x * 256 + threadIdx.x) * 4;
  const size_t n = idx & (size_t)(N - 1);
  const size_t d = (idx >> 11) & (size_t)(H - 1);
  const size_t b = idx >> 19;             // H*N = 2^19
  float4 mv = *(float4*)(Mem + idx);
  const float4 zv = *(const float4*)(Z + b * N + n);
  const float hw = Hw[b * H + d];
  mv.x = mv.x * (1.0f - zv.x) + hw * zv.x;
  mv.y = mv.y * (1.0f - zv.y) + hw * zv.y;
  mv.z = mv.z * (1.0f - zv.z) + hw * zv.z;
  mv.w = mv.w * (1.0f - zv.w) + hw * zv.w;
  *(float4*)(Mem + idx) = mv;
}

// ---------------------------------------------------------------------------
// Init / copy helpers
// ---------------------------------------------------------------------------
__global__ __launch_bounds__(256)
void init_mem_kernel(float* __restrict__ Mem, const float* __restrict__ MemBias)
{
  const size_t idx = ((size_t)blockIdx.x * 256 + threadIdx.x) * 4;
  const size_t dn = idx & ((size_t)H * N - 1);
  *(float4*)(Mem + idx) = *(const float4*)(MemBias + dn);
}

__global__ __launch_bounds__(256)
void init_state_kernel(float* __restrict__ hr0, float* __restrict__ cr,
                       float* __restrict__ hw0, float* __restrict__ cw,
                       const float* __restrict__ hid_h,
                       const float* __restrict__ hid_c)
{
  const int i = blockIdx.x * 256 + threadIdx.x;   // B*H
  hr0[i] = 0.0f;
  cr[i]  = 0.0f;
  hw0[i] = hid_h[i];
  cw[i]  = hid_c[i];
}

__global__ __launch_bounds__(256)
void copy_kernel(float* __restrict__ dst, const float* __restrict__ src)
{
  const int i = blockIdx.x * 256 + threadIdx.x;
  dst[i] = src[i];
}

// ---------------------------------------------------------------------------
extern "C" void kernel_launch(void* const* d_in, const int* in_sizes, int n_in,
                              void* d_out, int out_size, void* d_ws, size_t ws_size,
                              hipStream_t stream)
{
  const float* inp       = (const float*)d_in[0];   // [T,B,H]
  const float* hid_h     = (const float*)d_in[1];
  const float* hid_c     = (const float*)d_in[2];
  const float* wi_r      = (const float*)d_in[3];
  const float* wh_r      = (const float*)d_in[4];
  const float* bi_r      = (const float*)d_in[5];
  const float* bh_r      = (const float*)d_in[6];
  const float* wi_w      = (const float*)d_in[7];
  const float* wh_w      = (const float*)d_in[8];
  const float* bi_w      = (const float*)d_in[9];
  const float* bh_w      = (const float*)d_in[10];
  const float* compose_w = (const float*)d_in[11];  // [H,2H]
  const float* compose_b = (const float*)d_in[12];  // [H]
  const float* mem_bias  = (const float*)d_in[13];  // [H,N]

  // Output layout (flat, return order): out, h_w, c_w, mem, h_r, c_r
  float* out_seq = (float*)d_out;
  float* o_hw    = out_seq + (size_t)TSTEPS * B * H;
  float* o_cw    = o_hw + (size_t)B * H;
  float* mem     = o_cw + (size_t)B * H;
  float* o_hr    = mem + (size_t)B * H * N;
  float* o_cr    = o_hr + (size_t)B * H;

  // Workspace layout
  float* z    = (float*)d_ws;               // [B,N]
  float* mret = z    + (size_t)B * N;       // [B,H]
  float* ct   = mret + (size_t)B * H;
  float* hr0  = ct   + (size_t)B * H;
  float* hr1  = hr0  + (size_t)B * H;
  float* hw0  = hr1  + (size_t)B * H;
  float* hw1  = hw0  + (size_t)B * H;
  float* zp   = hw1  + (size_t)B * H;       // [B,nchunk,N]

  const size_t base_floats = (size_t)B * N + 6 * (size_t)B * H;
  int nchunk = 8;
  if (ws_size < (base_floats + (size_t)B * 8 * N) * sizeof(float)) nchunk = 1;

  const int memBlocks = (B * H * N / 4) / 256;  // 32768
  init_mem_kernel<<<memBlocks, 256, 0, stream>>>(mem, mem_bias);
  init_state_kernel<<<(B * H) / 256, 256, 0, stream>>>(hr0, o_cr, hw0, o_cw,
                                                       hid_h, hid_c);

  const dim3 gGemm(B / 16, H / 16);       // 64 waves
  const dim3 gAttn(B, nchunk);
  const dim3 gRetr(B, 4);

  for (int t = 0; t < TSTEPS; ++t) {
    float* hr_in  = (t & 1) ? hr1 : hr0;
    float* hr_out = (t & 1) ? hr0 : hr1;
    float* hw_in  = (t & 1) ? hw1 : hw0;
    float* hw_out = (t & 1) ? hw0 : hw1;

    lstm_cell_kernel<<<gGemm, 32, 0, stream>>>(
        inp + (size_t)t * B * H, wi_r, wh_r, bi_r, bh_r,
        hr_in, o_cr, hr_out, nullptr);

    attn_partial_kernel<<<gAttn, 256, 0, stream>>>(hr_out, mem, zp, nchunk);
    attn_finalize_kernel<<<B, 256, 0, stream>>>(zp, z, nchunk);
    retrieve_kernel<<<gRetr, 256, 0, stream>>>(mem, z, mret);

    compose_kernel<<<gGemm, 32, 0, stream>>>(hr_out, mret, compose_w,
                                             compose_b, ct);

    lstm_cell_kernel<<<gGemm, 32, 0, stream>>>(
        ct, wi_w, wh_w, bi_w, bh_w,
        hw_in, o_cw, hw_out, out_seq + (size_t)t * B * H);

    mem_update_kernel<<<memBlocks, 256, 0, stream>>>(mem, z, hw_out);
  }

  // After t = 15 (odd), final states are in the ping (index 0) buffers.
  copy_kernel<<<(B * H) / 256, 256, 0, stream>>>(o_hr, hr0);
  copy_kernel<<<(B * H) / 256, 256, 0, stream>>>(o_hw, hw0);
}